// NodeUpdateAttn_64750926954665
// MI455X (gfx1250) — compile-verified
//
#include <hip/hip_runtime.h>
#include <limits.h>

#define DCH 64
#define NEG_SLOPE 0.2f
#define EPS_IN 1e-5f

typedef float v2f __attribute__((ext_vector_type(2)));
typedef float v8f __attribute__((ext_vector_type(8)));

// ---------- native f32 atomics via CDNA5 inline asm (avoid any CAS-loop fallback) ----------
__device__ __forceinline__ void atomAddF(float* p, float v) {
  asm volatile("global_atomic_add_f32 %0, %1, off scope:SCOPE_DEV"
               :: "v"(p), "v"(v) : "memory");
}
__device__ __forceinline__ void atomMaxF(float* p, float v) {
  asm volatile("global_atomic_max_num_f32 %0, %1, off scope:SCOPE_DEV"
               :: "v"(p), "v"(v) : "memory");
}

// ---------- init kernels ----------
__global__ void init_layer1(float* __restrict__ acc, const float* __restrict__ bo,
                            float* __restrict__ smax, float* __restrict__ denom, int N) {
  int i = blockIdx.x * blockDim.x + threadIdx.x;
  if (i < N * DCH) acc[i] = bo[i & (DCH - 1)];
  if (i < N) { smax[i] = -__builtin_inff(); denom[i] = 0.f; }
}

__global__ void init_seg(float* __restrict__ smax, float* __restrict__ denom, int N) {
  int i = blockIdx.x * blockDim.x + threadIdx.x;
  if (i < N) { smax[i] = -__builtin_inff(); denom[i] = 0.f; }
}

// ---------- dense GEMM: C[M,64] = A[M,64] @ W^T + b1 (+ b2) ----------
// W row-major [n][k]; WMMA f32 16x16x4, one wave -> 16-row strip (4 n-tiles).
__global__ void gemm_node(const float* __restrict__ A, const float* __restrict__ W,
                          const float* __restrict__ b1, const float* __restrict__ b2,
                          float* __restrict__ C, int M) {
  int wave = blockIdx.x * (blockDim.x >> 5) + (threadIdx.x >> 5);
  int lane = threadIdx.x & 31;
  int m0 = wave * 16;
  if (m0 >= M) return;                       // whole-wave uniform
  int r = lane & 15;
  int hi = lane >> 4;

  v8f acc0 = {0,0,0,0,0,0,0,0};
  v8f acc1 = acc0, acc2 = acc0, acc3 = acc0;

  const float* Arow = A + (size_t)(m0 + r) * DCH;
#pragma unroll
  for (int k0 = 0; k0 < DCH; k0 += 4) {
    int kb = k0 + 2 * hi;
    v2f a; a.x = Arow[kb]; a.y = Arow[kb + 1];
    v2f b;
    b.x = W[(0 * 16 + r) * DCH + kb]; b.y = W[(0 * 16 + r) * DCH + kb + 1];
    acc0 = __builtin_amdgcn_wmma_f32_16x16x4_f32(false, a, false, b, (short)0, acc0, false, false);
    b.x = W[(1 * 16 + r) * DCH + kb]; b.y = W[(1 * 16 + r) * DCH + kb + 1];
    acc1 = __builtin_amdgcn_wmma_f32_16x16x4_f32(false, a, false, b, (short)0, acc1, false, false);
    b.x = W[(2 * 16 + r) * DCH + kb]; b.y = W[(2 * 16 + r) * DCH + kb + 1];
    acc2 = __builtin_amdgcn_wmma_f32_16x16x4_f32(false, a, false, b, (short)0, acc2, false, false);
    b.x = W[(3 * 16 + r) * DCH + kb]; b.y = W[(3 * 16 + r) * DCH + kb + 1];
    acc3 = __builtin_amdgcn_wmma_f32_16x16x4_f32(false, a, false, b, (short)0, acc3, false, false);
  }

  v8f accs[4] = {acc0, acc1, acc2, acc3};
#pragma unroll
  for (int t = 0; t < 4; ++t) {
    int n = t * 16 + r;
    float bias = b1[n] + (b2 ? b2[n] : 0.f);
#pragma unroll
    for (int v = 0; v < 8; ++v) {
      int m = m0 + v + 8 * hi;
      C[(size_t)m * DCH + n] = accs[t][v] + bias;
    }
  }
}

// ---------- fused edge kernel: e = edge_attr@We^T (WMMA), m = e + xl[src] + xr[dst],
// score = leaky(m).att ; atomic max_num into per-dst segment max. One wave = 16 edges.
__global__ void edge_score(const float* __restrict__ EA, const float* __restrict__ We,
                           const float* __restrict__ att,
                           const float* __restrict__ xl, const float* __restrict__ xr,
                           const long long* __restrict__ ei,
                           float* __restrict__ score, float* __restrict__ smax, int E) {
  int wave = blockIdx.x * (blockDim.x >> 5) + (threadIdx.x >> 5);
  int lane = threadIdx.x & 31;
  int e0 = wave * 16;
  if (e0 >= E) return;
  int r = lane & 15;
  int hi = lane >> 4;

  v8f acc0 = {0,0,0,0,0,0,0,0};
  v8f acc1 = acc0, acc2 = acc0, acc3 = acc0;

  const float* Arow = EA + (size_t)(e0 + r) * DCH;
#pragma unroll
  for (int k0 = 0; k0 < DCH; k0 += 4) {
    int kb = k0 + 2 * hi;
    v2f a; a.x = Arow[kb]; a.y = Arow[kb + 1];
    v2f b;
    b.x = We[(0 * 16 + r) * DCH + kb]; b.y = We[(0 * 16 + r) * DCH + kb + 1];
    acc0 = __builtin_amdgcn_wmma_f32_16x16x4_f32(false, a, false, b, (short)0, acc0, false, false);
    b.x = We[(1 * 16 + r) * DCH + kb]; b.y = We[(1 * 16 + r) * DCH + kb + 1];
    acc1 = __builtin_amdgcn_wmma_f32_16x16x4_f32(false, a, false, b, (short)0, acc1, false, false);
    b.x = We[(2 * 16 + r) * DCH + kb]; b.y = We[(2 * 16 + r) * DCH + kb + 1];
    acc2 = __builtin_amdgcn_wmma_f32_16x16x4_f32(false, a, false, b, (short)0, acc2, false, false);
    b.x = We[(3 * 16 + r) * DCH + kb]; b.y = We[(3 * 16 + r) * DCH + kb + 1];
    acc3 = __builtin_amdgcn_wmma_f32_16x16x4_f32(false, a, false, b, (short)0, acc3, false, false);
  }

  // per-lane edge indices for the 8 rows this lane's C registers touch
  long long sidx[8], didx[8];
#pragma unroll
  for (int v = 0; v < 8; ++v) {
    int m = e0 + v + 8 * hi;
    sidx[v] = ei[m];
    didx[v] = ei[(size_t)E + m];
  }

  float ps[8];
#pragma unroll
  for (int v = 0; v < 8; ++v) ps[v] = 0.f;

  v8f accs[4] = {acc0, acc1, acc2, acc3};
#pragma unroll
  for (int t = 0; t < 4; ++t) {
    int n = t * 16 + r;
    float av = att[n];
#pragma unroll
    for (int v = 0; v < 8; ++v) {
      float mval = accs[t][v] + xl[(size_t)sidx[v] * DCH + n] + xr[(size_t)didx[v] * DCH + n];
      mval = mval > 0.f ? mval : NEG_SLOPE * mval;
      ps[v] += mval * av;
    }
  }

  // butterfly reduce over the 16-lane half (channels); xor masks < 16 stay in-half
#pragma unroll
  for (int off = 8; off >= 1; off >>= 1) {
#pragma unroll
    for (int v = 0; v < 8; ++v) ps[v] += __shfl_xor(ps[v], off, 32);
  }

  if (r == 0) {  // lane 0 -> edges e0..e0+7, lane 16 -> e0+8..e0+15
#pragma unroll
    for (int v = 0; v < 8; ++v) {
      int m = e0 + v + 8 * hi;
      score[m] = ps[v];
      atomMaxF(&smax[(int)didx[v]], ps[v]);
    }
  }
}

// ---------- exp + denom ----------
__global__ void edge_exp(const long long* __restrict__ ei, float* __restrict__ score,
                         const float* __restrict__ smax, float* __restrict__ denom, int E) {
  int e = blockIdx.x * blockDim.x + threadIdx.x;
  if (e >= E) return;
  long long d = ei[(size_t)E + e];
  float mx = smax[(int)d];                 // finite: node d has edge e
  float es = expf(score[e] - mx);
  score[e] = es;
  atomAddF(&denom[(int)d], es);
}

// ---------- scatter: acc[dst] += alpha * xl[src]; one wave per edge ----------
__global__ void edge_scatter(const long long* __restrict__ ei, const float* __restrict__ es,
                             const float* __restrict__ denom, const float* __restrict__ xl,
                             float* __restrict__ acc, int E) {
  int wave = blockIdx.x * (blockDim.x >> 5) + (threadIdx.x >> 5);
  int lane = threadIdx.x & 31;
  if (wave >= E) return;
  long long s = ei[wave];
  long long d = ei[(size_t)E + wave];
  float alpha = es[wave] / (denom[(int)d] + 1e-16f);
  const float* xs = xl + (size_t)s * DCH;
  float* ad = acc + (size_t)d * DCH;
  atomAddF(&ad[lane],      alpha * xs[lane]);
  atomAddF(&ad[lane + 32], alpha * xs[lane + 32]);
}

// ---------- relu + instance-norm; one wave per node (2 ch/lane, wave32) ----------
__global__ void finalize_in(const float* __restrict__ acc, float* __restrict__ out, int N) {
  int wave = blockIdx.x * (blockDim.x >> 5) + (threadIdx.x >> 5);
  int lane = threadIdx.x & 31;
  if (wave >= N) return;
  const float* row = acc + (size_t)wave * DCH;
  float a = fmaxf(row[lane], 0.f);
  float b = fmaxf(row[lane + 32], 0.f);
  float s = a + b;
  float q = a * a + b * b;
#pragma unroll
  for (int off = 16; off >= 1; off >>= 1) {
    s += __shfl_xor(s, off, 32);
    q += __shfl_xor(q, off, 32);
  }
  float mu = s * (1.f / DCH);
  float var = q * (1.f / DCH) - mu * mu;
  float inv = rsqrtf(var + EPS_IN);
  float* orow = out + (size_t)wave * DCH;
  orow[lane]      = (a - mu) * inv;
  orow[lane + 32] = (b - mu) * inv;
}

// ---------- launcher ----------
extern "C" void kernel_launch(void* const* d_in, const int* in_sizes, int n_in,
                              void* d_out, int out_size, void* d_ws, size_t ws_size,
                              hipStream_t stream) {
  const float*     x    = (const float*)d_in[0];
  const long long* ei   = (const long long*)d_in[1];
  const float*     ea   = (const float*)d_in[2];
  const float *Wl1 = (const float*)d_in[3],  *bl1 = (const float*)d_in[4];
  const float *Wr1 = (const float*)d_in[5],  *br1 = (const float*)d_in[6];
  const float *We1 = (const float*)d_in[7],  *att1 = (const float*)d_in[8], *bo1 = (const float*)d_in[9];
  const float *Wl2 = (const float*)d_in[10], *bl2 = (const float*)d_in[11];
  const float *Wr2 = (const float*)d_in[12], *br2 = (const float*)d_in[13];
  const float *We2 = (const float*)d_in[14], *att2 = (const float*)d_in[15], *bo2 = (const float*)d_in[16];
  const float *Wsk = (const float*)d_in[17], *bsk = (const float*)d_in[18];

  const int N = in_sizes[0] / DCH;
  const int E = in_sizes[2] / DCH;

  float* xl    = (float*)d_ws;                       // N*64
  float* xr    = xl + (size_t)N * DCH;               // N*64
  float* h     = xr + (size_t)N * DCH;               // N*64
  float* acc   = h  + (size_t)N * DCH;               // N*64
  float* score = acc + (size_t)N * DCH;              // E
  float* denom = score + (size_t)E;                  // N
  float* smax  = denom + (size_t)N;                  // N

  const int BLK = 128;                               // 4 waves
  int gemmN_blocks  = ((N / 16) + 3) / 4;
  int escore_blocks = ((E / 16) + 3) / 4;
  int exp_blocks    = (E + 255) / 256;
  int scat_blocks   = (E + 7) / 8;                   // 8 waves per 256-thread block
  int fin_blocks    = (N + 7) / 8;
  int init1_blocks  = (N * DCH + 255) / 256;
  int initS_blocks  = (N + 255) / 256;

  // ===== layer 1 =====
  init_layer1<<<init1_blocks, 256, 0, stream>>>(acc, bo1, smax, denom, N);
  gemm_node<<<gemmN_blocks, BLK, 0, stream>>>(x, Wl1, bl1, nullptr, xl, N);
  gemm_node<<<gemmN_blocks, BLK, 0, stream>>>(x, Wr1, br1, nullptr, xr, N);
  edge_score<<<escore_blocks, BLK, 0, stream>>>(ea, We1, att1, xl, xr, ei, score, smax, E);
  edge_exp<<<exp_blocks, 256, 0, stream>>>(ei, score, smax, denom, E);
  edge_scatter<<<scat_blocks, 256, 0, stream>>>(ei, score, denom, xl, acc, E);
  finalize_in<<<fin_blocks, 256, 0, stream>>>(acc, h, N);

  // ===== layer 2 =====
  gemm_node<<<gemmN_blocks, BLK, 0, stream>>>(x, Wsk, bsk, bo2, acc, N);  // acc = skip + bskip + bo2
  init_seg<<<initS_blocks, 256, 0, stream>>>(smax, denom, N);
  gemm_node<<<gemmN_blocks, BLK, 0, stream>>>(h, Wl2, bl2, nullptr, xl, N);
  gemm_node<<<gemmN_blocks, BLK, 0, stream>>>(h, Wr2, br2, nullptr, xr, N);
  edge_score<<<escore_blocks, BLK, 0, stream>>>(ea, We2, att2, xl, xr, ei, score, smax, E);
  edge_exp<<<exp_blocks, 256, 0, stream>>>(ei, score, smax, denom, E);
  edge_scatter<<<scat_blocks, 256, 0, stream>>>(ei, score, denom, xl, acc, E);
  finalize_in<<<fin_blocks, 256, 0, stream>>>(acc, (float*)d_out, N);
}